// CorticalColumnGroup_53704271069426
// MI455X (gfx1250) — compile-verified
//
#include <hip/hip_runtime.h>
#include <math.h>

typedef __attribute__((ext_vector_type(2))) float v2f;
typedef __attribute__((ext_vector_type(8))) float v8f;

#define G_     64
#define DCOL   8
#define DPCM   32
#define DHID   32
#define NPOST  33
#define TOKENS 2048
#define TILE   16
#define NTILES (TOKENS / TILE)

// flat f32 output offsets (reference tuple order)
static constexpr size_t S8  = (size_t)TOKENS * G_ * DCOL;   // 1,048,576
static constexpr size_t S32 = (size_t)TOKENS * G_ * DPCM;   // 4,194,304
static constexpr size_t S1  = (size_t)TOKENS * G_;          //   131,072
static constexpr size_t O_XOUT = 0;
static constexpr size_t O_Z    = O_XOUT + S8;
static constexpr size_t O_ZHAT = O_Z + S32;
static constexpr size_t O_SUR  = O_ZHAT + S32;
static constexpr size_t O_KC   = O_SUR + S1;
static constexpr size_t O_VP   = O_KC + S8;
static constexpr size_t O_GATE = O_VP + S8;
static constexpr size_t O_QN   = O_GATE + S1;
static constexpr size_t O_VC   = O_QN + S8;
static constexpr size_t O_WN   = O_VC + S8;
static constexpr size_t O_S5   = O_WN + S1;

__device__ __forceinline__ v8f wmma4(v2f a, v2f b, v8f c) {
    return __builtin_amdgcn_wmma_f32_16x16x4_f32(
        /*neg_a=*/false, a, /*neg_b=*/false, b,
        /*c_mod=*/(short)0, c, /*reuse_a=*/false, /*reuse_b=*/false);
}

// D-frag(16x16) = A_lds(16xK) * Wg(K x ncols)[:, n0:n0+16] + bias, zero-padded cols.
template<int K>
__device__ __forceinline__ v8f mm_frag(const float* __restrict__ A_lds,
                                       const float* __restrict__ Wg, int ncols,
                                       const float* __restrict__ bg,
                                       int n0, int lane) {
    const int half = lane >> 4;
    const int nn   = lane & 15;
    const int row  = lane & 15;
    const int col  = n0 + nn;
    float bias = (col < ncols) ? bg[col] : 0.0f;
    v8f c;
#pragma unroll
    for (int r = 0; r < 8; ++r) c[r] = bias;
#pragma unroll
    for (int kk = 0; kk < K; kk += 4) {
        v2f a, b;
        a.x = A_lds[row * K + kk + 2 * half];
        a.y = A_lds[row * K + kk + 2 * half + 1];
        if (col < ncols) {
            b.x = Wg[(kk + 2 * half)     * ncols + col];
            b.y = Wg[(kk + 2 * half + 1) * ncols + col];
        } else {
            b.x = 0.0f; b.y = 0.0f;
        }
        c = wmma4(a, b, c);
    }
    return c;
}

__device__ __forceinline__ void store_frag(v8f c, float* __restrict__ Y,
                                           int ncols, int n0, int lane) {
    const int half = lane >> 4;
    const int nn   = lane & 15;
    const int col  = n0 + nn;
    if (col < ncols) {
#pragma unroll
        for (int r = 0; r < 8; ++r) Y[(r + 8 * half) * ncols + col] = c[r];
    }
}

__global__ __launch_bounds__(32)
void cortical_fused_kernel(const float* __restrict__ x_col,
                           const float* __restrict__ zhp,
                           const float* __restrict__ gamma,
                           const float* __restrict__ beta,
                           const float* __restrict__ W_up,   const float* __restrict__ b_up,
                           const float* __restrict__ W_down, const float* __restrict__ b_down,
                           const float* __restrict__ W_post, const float* __restrict__ b_post,
                           const float* __restrict__ W_enc,  const float* __restrict__ b_enc,
                           const float* __restrict__ W_pred, const float* __restrict__ b_pred,
                           const float* __restrict__ W_gain, const float* __restrict__ b_gain,
                           float* __restrict__ out) {
    __shared__ float Xs[TILE * DCOL];
    __shared__ float Zs[TILE * DPCM];
    __shared__ float Ds[TILE * DPCM];
    __shared__ float Gn[TILE * DCOL];
    __shared__ float Hs[TILE * DCOL];
    __shared__ float Us[TILE * DHID];
    __shared__ float HD[TILE * DCOL];
    __shared__ float XO[TILE * DCOL];
    __shared__ float PR[TILE * NPOST];

    const int lane = threadIdx.x;
    const int g    = blockIdx.x / NTILES;
    const int tile = blockIdx.x % NTILES;
    const int t0   = tile * TILE;
    const int half = lane >> 4;
    const int nn   = lane & 15;

    const float* Wenc  = W_enc  + (size_t)g * DCOL * DPCM;
    const float* Wgain = W_gain + (size_t)g * DPCM * DCOL;
    const float* Wup   = W_up   + (size_t)g * DCOL * DHID;
    const float* Wdown = W_down + (size_t)g * DHID * DCOL;
    const float* Wpred = W_pred + (size_t)g * DPCM * DPCM;
    const float* Wpost = W_post + (size_t)g * DCOL * NPOST;

    // 1) load x tile
    for (int i = lane; i < TILE * DCOL; i += 32) {
        int tl = i >> 3, d = i & 7;
        size_t T = (size_t)(t0 + tl);
        Xs[i] = x_col[(T * G_ + g) * DCOL + d];
    }
    __syncthreads();

    // 2) z = x @ W_enc + b_enc
    for (int nt = 0; nt < DPCM; nt += 16) {
        v8f c = mm_frag<DCOL>(Xs, Wenc, DPCM, b_enc + (size_t)g * DPCM, nt, lane);
        store_frag(c, Zs, DPCM, nt, lane);
    }
    __syncthreads();

    // 3) z out, delta = z - z_hat_prev
    for (int i = lane; i < TILE * DPCM; i += 32) {
        int tl = i >> 5, d = i & 31;
        size_t T = (size_t)(t0 + tl);
        float zv = Zs[i];
        out[O_Z + (T * G_ + g) * DPCM + d] = zv;
        Ds[i] = zv - zhp[(T * G_ + g) * DPCM + d];
    }
    __syncthreads();

    // 4) surprise / gate / surprise5 (per-token, lanes 0..15)
    if (lane < TILE) {
        float s2 = 0.0f;
#pragma unroll
        for (int d = 0; d < DPCM; ++d) { float v = Ds[lane * DPCM + d]; s2 += v * v; }
        float sur = sqrtf(s2);
        size_t T = (size_t)(t0 + lane);
        out[O_SUR  + T * G_ + g] = sur;
        out[O_GATE + T * G_ + g] = fminf(fmaxf(sur, 0.0f), 1.0f);
        out[O_S5   + T * G_ + g] = sur;
    }

    // 5) gain = 1 + 0.1*tanh(delta @ W_gain + b_gain)   (N=8, padded to 16)
    {
        v8f c = mm_frag<DPCM>(Ds, Wgain, DCOL, b_gain + (size_t)g * DCOL, 0, lane);
        if (nn < DCOL) {
#pragma unroll
            for (int r = 0; r < 8; ++r)
                Gn[(r + 8 * half) * DCOL + nn] = 1.0f + 0.1f * tanhf(c[r]);
        }
    }
    __syncthreads();

    // 6) LayerNorm * gamma + beta, * gain  (per-token, lanes 0..15)
    if (lane < TILE) {
        float mu = 0.0f;
#pragma unroll
        for (int d = 0; d < DCOL; ++d) mu += Xs[lane * DCOL + d];
        mu *= (1.0f / DCOL);
        float var = 0.0f;
#pragma unroll
        for (int d = 0; d < DCOL; ++d) { float v = Xs[lane * DCOL + d] - mu; var += v * v; }
        var *= (1.0f / DCOL);
        float inv = rsqrtf(var + 1e-5f);
#pragma unroll
        for (int d = 0; d < DCOL; ++d) {
            float h = (Xs[lane * DCOL + d] - mu) * inv * gamma[(size_t)g * DCOL + d]
                      + beta[(size_t)g * DCOL + d];
            Hs[lane * DCOL + d] = h * Gn[lane * DCOL + d];
        }
    }
    __syncthreads();

    // 7) u = gelu(h @ W_up + b_up)   (exact gelu)
    for (int nt = 0; nt < DHID; nt += 16) {
        v8f c = mm_frag<DCOL>(Hs, Wup, DHID, b_up + (size_t)g * DHID, nt, lane);
#pragma unroll
        for (int r = 0; r < 8; ++r) {
            float v = c[r];
            c[r] = 0.5f * v * (1.0f + erff(v * 0.70710678118654752f));
        }
        store_frag(c, Us, DHID, nt, lane);
    }
    __syncthreads();

    // 8) hd = u @ W_down + b_down   (N=8, padded)
    {
        v8f c = mm_frag<DHID>(Us, Wdown, DCOL, b_down + (size_t)g * DCOL, 0, lane);
        store_frag(c, HD, DCOL, 0, lane);
    }
    __syncthreads();

    // 9) x_out = x + hd
    for (int i = lane; i < TILE * DCOL; i += 32) {
        int tl = i >> 3, d = i & 7;
        size_t T = (size_t)(t0 + tl);
        float xo = Xs[i] + HD[i];
        XO[i] = xo;
        out[O_XOUT + (T * G_ + g) * DCOL + d] = xo;
    }
    __syncthreads();

    // 10) z_hat = z @ W_pred + b_pred   (stream D-frags straight to global)
    for (int nt = 0; nt < DPCM; nt += 16) {
        v8f c = mm_frag<DPCM>(Zs, Wpred, DPCM, b_pred + (size_t)g * DPCM, nt, lane);
#pragma unroll
        for (int r = 0; r < 8; ++r) {
            size_t T = (size_t)(t0 + r + 8 * half);
            out[O_ZHAT + (T * G_ + g) * DPCM + nt + nn] = c[r];
        }
    }

    // 11) proj = x_out @ W_post + b_post   (N=33 -> three padded tiles)
    for (int nt = 0; nt < 48; nt += 16) {
        v8f c = mm_frag<DCOL>(XO, Wpost, NPOST, b_post + (size_t)g * NPOST, nt, lane);
        store_frag(c, PR, NPOST, nt, lane);
    }
    __syncthreads();

    // 12) split proj, unit-normalize, sigmoid  (per-token, lanes 0..15)
    if (lane < TILE) {
        const float* p = PR + lane * NPOST;
        size_t T = (size_t)(t0 + lane);
        size_t base = T * G_ + g;
        float s = 0.0f;
#pragma unroll
        for (int d = 0; d < DCOL; ++d) s += p[d] * p[d];
        float inv1 = 1.0f / (sqrtf(s) + 1e-6f);
        s = 0.0f;
#pragma unroll
        for (int d = 0; d < DCOL; ++d) s += p[16 + d] * p[16 + d];
        float inv2 = 1.0f / (sqrtf(s) + 1e-6f);
#pragma unroll
        for (int d = 0; d < DCOL; ++d) {
            out[O_KC + base * DCOL + d] = p[d] * inv1;        // k_cand = norm(k_pre)
            out[O_VP + base * DCOL + d] = p[8 + d];           // v_post
            out[O_QN + base * DCOL + d] = p[16 + d] * inv2;   // q_nov = norm(k_cand_raw)
            out[O_VC + base * DCOL + d] = p[24 + d];          // v_cand
        }
        out[O_WN + base] = 1.0f / (1.0f + expf(-p[32]));      // w_nov
    }
}

extern "C" void kernel_launch(void* const* d_in, const int* in_sizes, int n_in,
                              void* d_out, int out_size, void* d_ws, size_t ws_size,
                              hipStream_t stream) {
    const float* x_col  = (const float*)d_in[0];
    const float* zhp    = (const float*)d_in[1];
    const float* gamma  = (const float*)d_in[2];
    const float* beta   = (const float*)d_in[3];
    const float* W_up   = (const float*)d_in[4];
    const float* b_up   = (const float*)d_in[5];
    const float* W_down = (const float*)d_in[6];
    const float* b_down = (const float*)d_in[7];
    const float* W_post = (const float*)d_in[8];
    const float* b_post = (const float*)d_in[9];
    const float* W_enc  = (const float*)d_in[10];
    const float* b_enc  = (const float*)d_in[11];
    const float* W_pred = (const float*)d_in[12];
    const float* b_pred = (const float*)d_in[13];
    const float* W_gain = (const float*)d_in[14];
    const float* b_gain = (const float*)d_in[15];
    float* out = (float*)d_out;

    dim3 grid(G_ * NTILES);   // 64 groups * 128 token tiles = 8192 waves
    dim3 block(32);           // one wave32 per block (WMMA needs full wave)
    cortical_fused_kernel<<<grid, block, 0, stream>>>(
        x_col, zhp, gamma, beta, W_up, b_up, W_down, b_down,
        W_post, b_post, W_enc, b_enc, W_pred, b_pred, W_gain, b_gain, out);
}